// EdgeConv_39496519254520
// MI455X (gfx1250) — compile-verified
//
#include <hip/hip_runtime.h>
#include <math.h>

typedef float v2f __attribute__((ext_vector_type(2)));
typedef float v8f __attribute__((ext_vector_type(8)));

#define D 64  // D_IN == D_OUT == 64

// ---------------------------------------------------------------------------
// prep: W2 = phi_w - theta_w  (the "src" weight), bsum = theta_b + phi_b
// ---------------------------------------------------------------------------
__global__ void ec_prep(const float* __restrict__ theta_w,
                        const float* __restrict__ theta_b,
                        const float* __restrict__ phi_w,
                        const float* __restrict__ phi_b,
                        float* __restrict__ W2,
                        float* __restrict__ bsum) {
    int i = blockIdx.x * blockDim.x + threadIdx.x;
    if (i < D * D) W2[i] = phi_w[i] - theta_w[i];
    if (i < D)     bsum[i] = theta_b[i] + phi_b[i];
}

// ---------------------------------------------------------------------------
// init: M[:] = -inf (scatter-max identity). Re-done every launch (graph replay).
// ---------------------------------------------------------------------------
__global__ void ec_init(float4* __restrict__ M4, int n4) {
    int i = blockIdx.x * blockDim.x + threadIdx.x;
    if (i < n4) M4[i] = make_float4(-INFINITY, -INFINITY, -INFINITY, -INFINITY);
}

// ---------------------------------------------------------------------------
// Dual node GEMM via V_WMMA_F32_16X16X4_F32:
//   A[n]  = x[n] @ theta_w
//   Bv[n] = x[n] @ (phi_w - theta_w) + (theta_b + phi_b)
// One block (8 waves) per 16-row node tile. Waves 0-3 -> A cols, 4-7 -> Bv cols.
// f32 WMMA fragment layout (wave32):
//   A 16x4 : lane l holds row (l&15), Ks {2*(l>>4), 2*(l>>4)+1} in 2 VGPRs
//   B 4x16 : lane l holds col (l&15), Ks {2*(l>>4), 2*(l>>4)+1} in 2 VGPRs
//   C 16x16: VGPR v, lane l -> (row v + 8*(l>>4), col l&15)
// ---------------------------------------------------------------------------
__global__ __launch_bounds__(256) void ec_gemm(const float* __restrict__ x,   // [N,64]
                                               const float* __restrict__ Wa,  // theta_w [64,64] (k-major rows)
                                               const float* __restrict__ W2,  // phi-theta [64,64]
                                               const float* __restrict__ bsum,// [64]
                                               float* __restrict__ A,         // [N,64]
                                               float* __restrict__ Bv) {      // [N,64]
    const int lane    = threadIdx.x & 31;
    const int wave    = threadIdx.x >> 5;
    const int rowbase = blockIdx.x * 16;
    const bool second = (wave >= 4);

    const float* Bmat = second ? W2 : Wa;
    float*       dest = second ? Bv : A;
    const int colbase = (wave & 3) * 16;

    const int r    = rowbase + (lane & 15);     // source row for A-fragment
    const int ksel = (lane >> 4) * 2;           // K offset within 4-chunk: 0 or 2
    const int n    = colbase + (lane & 15);     // output column for B/C fragment

    v8f acc = {0.f, 0.f, 0.f, 0.f, 0.f, 0.f, 0.f, 0.f};

#pragma unroll
    for (int kk = 0; kk < 16; ++kk) {           // K = 64 = 16 * 4
        const int k0 = kk * 4;
        v2f a = *(const v2f*)(x + r * D + k0 + ksel);
        v2f b;
        b.x = Bmat[(k0 + ksel)     * D + n];
        b.y = Bmat[(k0 + ksel + 1) * D + n];
        acc = __builtin_amdgcn_wmma_f32_16x16x4_f32(
            /*neg_a=*/false, a, /*neg_b=*/false, b,
            /*c_mod=*/(short)0, acc, /*reuse_a=*/false, /*reuse_b=*/false);
    }

    const float bval = second ? bsum[n] : 0.0f;
    const int   mrow = rowbase + ((lane >> 4) ? 8 : 0);
#pragma unroll
    for (int v = 0; v < 8; ++v) {
        dest[(mrow + v) * D + n] = acc[v] + bval;
    }
}

// ---------------------------------------------------------------------------
// Edge scatter-max: one wave per edge. Lane l gathers Bv[src][2l..2l+1]
// (coalesced 256B row) and atomically maxes into M[dst][2l..2l+1].
// Uses CDNA5 GLOBAL_ATOMIC_MAX_NUM_F32 (no return -> STOREcnt, order-free).
// ---------------------------------------------------------------------------
__global__ __launch_bounds__(256) void ec_scatter(const int* __restrict__ src,
                                                  const int* __restrict__ dst,
                                                  const float* __restrict__ Bv,
                                                  float* __restrict__ M,
                                                  int nEdges) {
    const int lane = threadIdx.x & 31;
    const int e    = (blockIdx.x * blockDim.x + threadIdx.x) >> 5;
    if (e >= nEdges) return;

    const int s = src[e];
    const int d = dst[e];

    v2f val = *(const v2f*)(Bv + (size_t)s * D + lane * 2);
    float* p = M + (size_t)d * D + lane * 2;

    asm volatile("global_atomic_max_num_f32 %0, %1, off"
                 :: "v"(p),     "v"(val.x) : "memory");
    asm volatile("global_atomic_max_num_f32 %0, %1, off"
                 :: "v"(p + 1), "v"(val.y) : "memory");
}

// ---------------------------------------------------------------------------
// combine: out = (M == -inf) ? 0 : A + M   (matches DGL fill of edgeless nodes)
// ---------------------------------------------------------------------------
__global__ void ec_combine(const float4* __restrict__ A4,
                           const float4* __restrict__ M4,
                           float4* __restrict__ out4, int n4) {
    int i = blockIdx.x * blockDim.x + threadIdx.x;
    if (i >= n4) return;
    float4 a = A4[i];
    float4 m = M4[i];
    float4 o;
    o.x = (m.x == -INFINITY) ? 0.0f : a.x + m.x;
    o.y = (m.y == -INFINITY) ? 0.0f : a.y + m.y;
    o.z = (m.z == -INFINITY) ? 0.0f : a.z + m.z;
    o.w = (m.w == -INFINITY) ? 0.0f : a.w + m.w;
    out4[i] = o;
}

// ---------------------------------------------------------------------------
extern "C" void kernel_launch(void* const* d_in, const int* in_sizes, int n_in,
                              void* d_out, int out_size, void* d_ws, size_t ws_size,
                              hipStream_t stream) {
    // inputs in setup_inputs() order
    const float* h       = (const float*)d_in[0];  // [B,P,64] -> [N,64]
    const int*   src     = (const int*)  d_in[1];  // [E]
    const int*   dst     = (const int*)  d_in[2];  // [E]
    const float* theta_w = (const float*)d_in[3];  // [64,64]
    const float* theta_b = (const float*)d_in[4];  // [64]
    const float* phi_w   = (const float*)d_in[5];  // [64,64]
    const float* phi_b   = (const float*)d_in[6];  // [64]
    float*       out     = (float*)d_out;          // [N,64]

    const int N = in_sizes[0] / D;   // 65536
    const int E = in_sizes[1];       // 1310720

    // workspace layout (floats), 16B-aligned slices; total ~48.3 MB
    float* wsf  = (float*)d_ws;
    float* W2   = wsf;                       // 64*64
    float* bsum = wsf + D * D;               // 64
    float* A    = wsf + D * D + 256;         // N*64
    float* Bv   = A + (size_t)N * D;         // N*64
    float* M    = Bv + (size_t)N * D;        // N*64

    const int n4 = (N * D) / 4;

    ec_prep<<<(D * D + 255) / 256, 256, 0, stream>>>(theta_w, theta_b, phi_w, phi_b, W2, bsum);
    ec_init<<<(n4 + 255) / 256, 256, 0, stream>>>((float4*)M, n4);
    ec_gemm<<<N / 16, 256, 0, stream>>>(h, theta_w, W2, bsum, A, Bv);
    ec_scatter<<<((size_t)E * 32 + 255) / 256, 256, 0, stream>>>(src, dst, Bv, M, E);
    ec_combine<<<(n4 + 255) / 256, 256, 0, stream>>>((const float4*)A, (const float4*)M,
                                                     (float4*)out, n4);
}